// minerva_ffnn4_62912680951927
// MI455X (gfx1250) — compile-verified
//
#include <hip/hip_runtime.h>
#include <math.h>

typedef __attribute__((ext_vector_type(2))) float v2f;
typedef __attribute__((ext_vector_type(8))) float v8f;

#define B_SZ 1024
#define D_SZ 50000
#define H_SZ 1024
#define F_SZ 256
#define C_SZ 28

#define LDP 66  // padded LDS row stride (floats): bank = 2*l mod 64, conflict-free

// ---------------- ex_class_reps = ex_classes @ class_reps  [D,C] ----------------
__global__ void __launch_bounds__(256) ecr_kernel(const float* __restrict__ ex_classes,
                                                  const float* __restrict__ class_reps,
                                                  float* __restrict__ ecr) {
    __shared__ float cr[C_SZ * C_SZ];
    int tid = threadIdx.x;
    for (int i = tid; i < C_SZ * C_SZ; i += 256) cr[i] = class_reps[i];
    __syncthreads();
    int idx = blockIdx.x * 256 + tid;
    if (idx >= D_SZ * C_SZ) return;
    int d = idx / C_SZ, c = idx % C_SZ;
    const float* ec = ex_classes + (size_t)d * C_SZ;
    float acc = 0.f;
#pragma unroll
    for (int j = 0; j < C_SZ; ++j) acc = fmaf(ec[j], cr[j * C_SZ + c], acc);
    ecr[idx] = acc;
}

// ------------- WMMA GEMM: Y[M,N] = X[M,K=1024] @ W[N,K=1024]^T + bias[N] ---------
// Block 256 threads = 8 waves (2x4). Wave tile 32x32 (2x2 WMMA subtiles),
// block tile 64(M) x 128(N). K chunked by 64 through LDS (stride-66 padding).
// grid.x = N/128 (=2), grid.y = ceil(M/64).
__global__ void __launch_bounds__(256) proj_gemm_kernel(const float* __restrict__ X,
                                                        const float* __restrict__ W,
                                                        const float* __restrict__ bias,
                                                        float* __restrict__ Y, int M) {
    __shared__ float xs[64 * LDP];
    __shared__ float ws[128 * LDP];
    int tid  = threadIdx.x;
    int wave = tid >> 5, lane = tid & 31;
    int l    = lane & 15, half = lane >> 4;
    int m0   = (wave >> 2) * 32;  // 0 or 32
    int nloc = (wave & 3) * 32;   // 0..96
    int mbase = blockIdx.y * 64;
    int nbase = blockIdx.x * 128;

    v8f acc00 = {}, acc01 = {}, acc10 = {}, acc11 = {};
    for (int kc = 0; kc < H_SZ; kc += 64) {
        // stage X tile: 64x64 floats = 1024 float4, 4/thread (OOB rows -> 0)
#pragma unroll
        for (int t = 0; t < 4; ++t) {
            int idx = tid + t * 256;
            int row = idx >> 4, c4 = (idx & 15) << 2;
            int gr = mbase + row;
            float4 v = make_float4(0.f, 0.f, 0.f, 0.f);
            if (gr < M) v = *(const float4*)(X + (size_t)gr * H_SZ + kc + c4);
            float* dst = &xs[row * LDP + c4];
            ((float2*)dst)[0] = make_float2(v.x, v.y);
            ((float2*)dst)[1] = make_float2(v.z, v.w);
        }
        // stage W tile: 128x64 floats = 2048 float4, 8/thread (N rows always valid)
#pragma unroll
        for (int t = 0; t < 8; ++t) {
            int idx = tid + t * 256;
            int row = idx >> 4, c4 = (idx & 15) << 2;
            float4 v = *(const float4*)(W + (size_t)(nbase + row) * H_SZ + kc + c4);
            float* dst = &ws[row * LDP + c4];
            ((float2*)dst)[0] = make_float2(v.x, v.y);
            ((float2*)dst)[1] = make_float2(v.z, v.w);
        }
        __syncthreads();
#pragma unroll
        for (int kk = 0; kk < 64; kk += 4) {
            int kb = kk + 2 * half;
            float2 a0 = *(const float2*)&xs[(m0 + l) * LDP + kb];
            float2 a1 = *(const float2*)&xs[(m0 + 16 + l) * LDP + kb];
            float2 b0 = *(const float2*)&ws[(nloc + l) * LDP + kb];
            float2 b1 = *(const float2*)&ws[(nloc + 16 + l) * LDP + kb];
            v2f av0; av0.x = a0.x; av0.y = a0.y;
            v2f av1; av1.x = a1.x; av1.y = a1.y;
            v2f bv0; bv0.x = b0.x; bv0.y = b0.y;
            v2f bv1; bv1.x = b1.x; bv1.y = b1.y;
            acc00 = __builtin_amdgcn_wmma_f32_16x16x4_f32(false, av0, false, bv0, (short)0, acc00, false, false);
            acc01 = __builtin_amdgcn_wmma_f32_16x16x4_f32(false, av0, false, bv1, (short)0, acc01, false, false);
            acc10 = __builtin_amdgcn_wmma_f32_16x16x4_f32(false, av1, false, bv0, (short)0, acc10, false, false);
            acc11 = __builtin_amdgcn_wmma_f32_16x16x4_f32(false, av1, false, bv1, (short)0, acc11, false, false);
        }
        __syncthreads();
    }
    int col0 = nbase + nloc + l;
    int col1 = col0 + 16;
    float bi0 = bias[col0], bi1 = bias[col1];
#pragma unroll
    for (int i = 0; i < 8; ++i) {
        int r0 = mbase + m0 + i + 8 * half;
        int r1 = r0 + 16;
        if (r0 < M) {
            Y[(size_t)r0 * F_SZ + col0] = acc00[i] + bi0;
            Y[(size_t)r0 * F_SZ + col1] = acc01[i] + bi1;
        }
        if (r1 < M) {
            Y[(size_t)r1 * F_SZ + col0] = acc10[i] + bi0;
            Y[(size_t)r1 * F_SZ + col1] = acc11[i] + bi1;
        }
    }
}

// ------------- in-place row L2 normalization, rows of 256 floats -----------------
// One wave per row, 8 rows per block.
__global__ void __launch_bounds__(256) norm_rows_kernel(float* __restrict__ Y, int R) {
    int tid = threadIdx.x;
    int wave = tid >> 5, lane = tid & 31;
    int row = blockIdx.x * 8 + wave;
    if (row >= R) return;  // whole wave exits uniformly
    float4* p = (float4*)(Y + (size_t)row * F_SZ);
    float4 v0 = p[lane], v1 = p[lane + 32];
    float s = v0.x * v0.x + v0.y * v0.y + v0.z * v0.z + v0.w * v0.w +
              v1.x * v1.x + v1.y * v1.y + v1.z * v1.z + v1.w * v1.w;
#pragma unroll
    for (int o = 16; o > 0; o >>= 1) s += __shfl_xor(s, o, 32);
    float inv = 1.f / fmaxf(sqrtf(s), 1e-12f);
    v0.x *= inv; v0.y *= inv; v0.z *= inv; v0.w *= inv;
    v1.x *= inv; v1.y *= inv; v1.z *= inv; v1.w *= inv;
    p[lane] = v0; p[lane + 32] = v1;
}

// ------------- a = (qn @ kn^T)^3 via V_WMMA_F32_16X16X4_F32 ----------------------
// Same macro-tiling: block tile 64(B) x 128(D), wave tile 32x32, K=256 in 4 chunks.
__global__ void __launch_bounds__(256) sim_pow_kernel(const float* __restrict__ qn,
                                                      const float* __restrict__ kn,
                                                      float* __restrict__ aout) {
    __shared__ float qs[64 * LDP];
    __shared__ float ks[128 * LDP];
    int tid  = threadIdx.x;
    int wave = tid >> 5, lane = tid & 31;
    int l    = lane & 15, half = lane >> 4;
    int m0   = (wave >> 2) * 32;
    int nloc = (wave & 3) * 32;
    int b0 = blockIdx.y * 64;       // B rows: 1024 % 64 == 0, no guard
    int n0 = blockIdx.x * 128;      // D cols: guard

    v8f acc00 = {}, acc01 = {}, acc10 = {}, acc11 = {};
    for (int kc = 0; kc < F_SZ; kc += 64) {
        // stage q tile: 64x64 = 1024 float4, 4/thread
#pragma unroll
        for (int t = 0; t < 4; ++t) {
            int idx = tid + t * 256;
            int row = idx >> 4, c4 = (idx & 15) << 2;
            float4 v = *(const float4*)(qn + (size_t)(b0 + row) * F_SZ + kc + c4);
            float* dst = &qs[row * LDP + c4];
            ((float2*)dst)[0] = make_float2(v.x, v.y);
            ((float2*)dst)[1] = make_float2(v.z, v.w);
        }
        // stage k tile: 128x64 = 2048 float4, 8/thread (OOB rows -> 0)
#pragma unroll
        for (int t = 0; t < 8; ++t) {
            int idx = tid + t * 256;
            int row = idx >> 4, c4 = (idx & 15) << 2;
            int gr = n0 + row;
            float4 v = make_float4(0.f, 0.f, 0.f, 0.f);
            if (gr < D_SZ) v = *(const float4*)(kn + (size_t)gr * F_SZ + kc + c4);
            float* dst = &ks[row * LDP + c4];
            ((float2*)dst)[0] = make_float2(v.x, v.y);
            ((float2*)dst)[1] = make_float2(v.z, v.w);
        }
        __syncthreads();
#pragma unroll
        for (int kk = 0; kk < 64; kk += 4) {
            int kb = kk + 2 * half;
            float2 a0 = *(const float2*)&qs[(m0 + l) * LDP + kb];
            float2 a1 = *(const float2*)&qs[(m0 + 16 + l) * LDP + kb];
            float2 b0v = *(const float2*)&ks[(nloc + l) * LDP + kb];
            float2 b1v = *(const float2*)&ks[(nloc + 16 + l) * LDP + kb];
            v2f av0; av0.x = a0.x; av0.y = a0.y;
            v2f av1; av1.x = a1.x; av1.y = a1.y;
            v2f bv0; bv0.x = b0v.x; bv0.y = b0v.y;
            v2f bv1; bv1.x = b1v.x; bv1.y = b1v.y;
            acc00 = __builtin_amdgcn_wmma_f32_16x16x4_f32(false, av0, false, bv0, (short)0, acc00, false, false);
            acc01 = __builtin_amdgcn_wmma_f32_16x16x4_f32(false, av0, false, bv1, (short)0, acc01, false, false);
            acc10 = __builtin_amdgcn_wmma_f32_16x16x4_f32(false, av1, false, bv0, (short)0, acc10, false, false);
            acc11 = __builtin_amdgcn_wmma_f32_16x16x4_f32(false, av1, false, bv1, (short)0, acc11, false, false);
        }
        __syncthreads();
    }
    // a = s^3; C/D layout: VGPR i -> M = i + 8*half, N = l
    int col0 = n0 + nloc + l;
    int col1 = col0 + 16;
#pragma unroll
    for (int i = 0; i < 8; ++i) {
        int r0 = b0 + m0 + i + 8 * half;
        int r1 = r0 + 16;
        float s;
        if (col0 < D_SZ) {
            s = acc00[i]; aout[(size_t)r0 * D_SZ + col0] = s * s * s;
            s = acc10[i]; aout[(size_t)r1 * D_SZ + col0] = s * s * s;
        }
        if (col1 < D_SZ) {
            s = acc01[i]; aout[(size_t)r0 * D_SZ + col1] = s * s * s;
            s = acc11[i]; aout[(size_t)r1 * D_SZ + col1] = s * s * s;
        }
    }
}

// ------------- echo = a @ ecr ; logits = (echo @ class_reps^T)*mult - thresh -----
__global__ void __launch_bounds__(256) echo_kernel(const float* __restrict__ a,
                                                   const float* __restrict__ ecr,
                                                   const float* __restrict__ class_reps,
                                                   const float* __restrict__ mult,
                                                   const float* __restrict__ thresh,
                                                   float* __restrict__ echo_out,
                                                   float* __restrict__ logits_out) {
    __shared__ float part[8][C_SZ];
    int b = blockIdx.x, tid = threadIdx.x;
    int wave = tid >> 5, lane = tid & 31;
    float acc[C_SZ];
#pragma unroll
    for (int c = 0; c < C_SZ; ++c) acc[c] = 0.f;
    const float* arow = a + (size_t)b * D_SZ;
    for (int d = tid; d < D_SZ; d += 256) {
        float av = arow[d];
        const float4* e4 = (const float4*)(ecr + (size_t)d * C_SZ);
#pragma unroll
        for (int q = 0; q < C_SZ / 4; ++q) {
            float4 e = e4[q];
            acc[q * 4 + 0] = fmaf(av, e.x, acc[q * 4 + 0]);
            acc[q * 4 + 1] = fmaf(av, e.y, acc[q * 4 + 1]);
            acc[q * 4 + 2] = fmaf(av, e.z, acc[q * 4 + 2]);
            acc[q * 4 + 3] = fmaf(av, e.w, acc[q * 4 + 3]);
        }
    }
#pragma unroll
    for (int c = 0; c < C_SZ; ++c) {
        float v = acc[c];
        for (int o = 16; o > 0; o >>= 1) v += __shfl_down(v, o, 32);
        if (lane == 0) part[wave][c] = v;
    }
    __syncthreads();
    if (tid < C_SZ) {
        float e = 0.f;
#pragma unroll
        for (int w = 0; w < 8; ++w) e += part[w][tid];
        part[0][tid] = e;
    }
    __syncthreads();
    if (tid < C_SZ) {
        float e = part[0][tid];
        echo_out[(size_t)b * C_SZ + tid] = e;
        float lg = 0.f;
#pragma unroll
        for (int j = 0; j < C_SZ; ++j) lg = fmaf(part[0][j], class_reps[tid * C_SZ + j], lg);
        logits_out[(size_t)b * C_SZ + tid] = lg * mult[tid] - thresh[tid];
    }
}

extern "C" void kernel_launch(void* const* d_in, const int* in_sizes, int n_in,
                              void* d_out, int out_size, void* d_ws, size_t ws_size,
                              hipStream_t stream) {
    const float* features    = (const float*)d_in[0];
    const float* ex_features = (const float*)d_in[1];
    const float* ex_classes  = (const float*)d_in[2];
    const float* class_reps  = (const float*)d_in[3];
    const float* Wq = (const float*)d_in[4];
    const float* bq = (const float*)d_in[5];
    const float* Wk = (const float*)d_in[6];
    const float* bk = (const float*)d_in[7];
    const float* mult   = (const float*)d_in[8];
    const float* thresh = (const float*)d_in[9];

    float* out        = (float*)d_out;
    float* echo_out   = out;                              // [B,C]
    float* logits_out = out + (size_t)B_SZ * C_SZ;        // [B,C]
    float* a_out      = out + 2 * (size_t)B_SZ * C_SZ;    // [B,D]

    float* kn  = (float*)d_ws;                            // D*F floats (51.2 MB)
    float* qn  = kn + (size_t)D_SZ * F_SZ;                // B*F floats
    float* ecr = qn + (size_t)B_SZ * F_SZ;                // D*C floats

    ecr_kernel<<<(D_SZ * C_SZ + 255) / 256, 256, 0, stream>>>(ex_classes, class_reps, ecr);

    // projections on the fp32 WMMA pipe
    dim3 gridQ(F_SZ / 128, (B_SZ + 63) / 64);
    proj_gemm_kernel<<<gridQ, 256, 0, stream>>>(features, Wq, bq, qn, B_SZ);
    dim3 gridK(F_SZ / 128, (D_SZ + 63) / 64);
    proj_gemm_kernel<<<gridK, 256, 0, stream>>>(ex_features, Wk, bk, kn, D_SZ);

    norm_rows_kernel<<<(B_SZ + 7) / 8, 256, 0, stream>>>(qn, B_SZ);
    norm_rows_kernel<<<(D_SZ + 7) / 8, 256, 0, stream>>>(kn, D_SZ);

    dim3 gridS((D_SZ + 127) / 128, B_SZ / 64);
    sim_pow_kernel<<<gridS, 256, 0, stream>>>(qn, kn, a_out);

    echo_kernel<<<B_SZ, 256, 0, stream>>>(a_out, ecr, class_reps, mult, thresh,
                                          echo_out, logits_out);
}